// MeshRCNNGraphConvHead_28406913696540
// MI455X (gfx1250) — compile-verified
//
#include <hip/hip_runtime.h>
#include <math.h>

// Problem constants (match reference)
#define VN  160000
#define EN  480000
#define NB  16
#define CC  256
#define HWD 14
#define HID 128

typedef float v2f __attribute__((ext_vector_type(2)));
typedef float v8f __attribute__((ext_vector_type(8)));

// ---------------------------------------------------------------------------
// x (N,C,H,W) -> xT (N,H,W,C) so per-vertex channel gathers are coalesced
__global__ void transpose_x_kernel(const float* __restrict__ x, float* __restrict__ xT) {
    int idx = blockIdx.x * blockDim.x + threadIdx.x;     // (b*196+pix)*256 + c
    if (idx >= NB * CC * HWD * HWD) return;
    int c    = idx & (CC - 1);
    int rest = idx >> 8;                                  // b*196 + pix
    int pix  = rest % (HWD * HWD);
    int b    = rest / (HWD * HWD);
    xT[idx] = x[((size_t)b * CC + c) * (HWD * HWD) + pix];
}

__global__ void init_pos_kernel(const float* __restrict__ verts, float* __restrict__ vcur) {
    int idx = blockIdx.x * blockDim.x + threadIdx.x;
    if (idx < VN * 3) vcur[idx] = verts[idx];
}

// Zero-pad weight matrix (K x 128) to (Kpad x 128): pad rows are 0 so the GEMM
// inner loop needs no bounds checks (no divergent EXEC masking around WMMA).
__global__ void pad_w_kernel(const float* __restrict__ W, float* __restrict__ Wp,
                             int K, int Kpad) {
    int idx = blockIdx.x * blockDim.x + threadIdx.x;     // r*128 + c
    if (idx >= Kpad * HID) return;
    int r = idx >> 7;
    Wp[idx] = (r < K) ? W[idx] : 0.f;
}

// ---------------------------------------------------------------------------
// Bilinear vert_align: S[v][c] = bilerp(xT[b], gx, gy)[c]
__global__ void sample_kernel(const float* __restrict__ xT, const float* __restrict__ vcur,
                              const int* __restrict__ v2m, float* __restrict__ S) {
    int idx = blockIdx.x * blockDim.x + threadIdx.x;     // v*256 + c
    int v = idx >> 8;
    int c = idx & 255;
    if (v >= VN) return;
    int b = v2m[v];
    float vx = vcur[v * 3 + 0], vy = vcur[v * 3 + 1];
    float gx = fminf(fmaxf((vx + 1.f) * 0.5f * (HWD - 1), 0.f), (float)(HWD - 1));
    float gy = fminf(fmaxf((vy + 1.f) * 0.5f * (HWD - 1), 0.f), (float)(HWD - 1));
    float x0 = floorf(gx), y0 = floorf(gy);
    float wx = gx - x0,    wy = gy - y0;
    int x0i = (int)x0, y0i = (int)y0;
    int x1i = x0i + 1 < HWD - 1 ? x0i + 1 : HWD - 1;
    int y1i = y0i + 1 < HWD - 1 ? y0i + 1 : HWD - 1;
    const float* base = xT + (size_t)b * (HWD * HWD * CC);
    float f00 = base[(y0i * HWD + x0i) * CC + c];
    float f01 = base[(y0i * HWD + x1i) * CC + c];
    float f10 = base[(y1i * HWD + x0i) * CC + c];
    float f11 = base[(y1i * HWD + x1i) * CC + c];
    S[idx] = (1.f - wx) * (1.f - wy) * f00 + wx * (1.f - wy) * f01
           + (1.f - wx) * wy * f10        + wx * wy * f11;
}

// ---------------------------------------------------------------------------
// fp32 WMMA GEMM: out[v0:v0+32, ocol:ocol+128] = A[v0:v0+32, 0:K] @ Wp + bias
// Block = 256 threads = 8 waves. Wave j owns N-tile j (16 cols) for BOTH 16-row
// M-subtiles (B fragment reused x2). A tile staged zero-padded in LDS; Wp is
// zero-padded in global, so the K loop is branch-free (EXEC stays all-1s).
__global__ __launch_bounds__(256) void gemm_wmma_kernel(
    const float* __restrict__ A, int lda, int K,
    const float* __restrict__ Wp,    // Kpad x 128, zero-padded rows
    const float* __restrict__ bias,
    float* __restrict__ out, int ldo, int ocol, int relu)
{
    __shared__ float As[32 * 260];
    int Kpad = (K + 3) & ~3;
    int v0   = blockIdx.x * 32;
    int tid  = threadIdx.x;
    int wave = tid >> 5, lane = tid & 31;

    // Stage A tile (32 rows x Kpad), coalesced, zero-pad cols >= K
    for (int rr = 0; rr < 4; rr++) {
        int r = wave * 4 + rr;
        const float* arow = A + (size_t)(v0 + r) * lda;
        float* drow = As + r * 260;
        for (int c = lane; c < Kpad; c += 32)
            drow[c] = (c < K) ? arow[c] : 0.f;
    }
    __syncthreads();

    int nl = lane & 15;          // N index within tile; also M row for A frag
    int kh = lane >> 4;          // K-half (A/B frags), M-half (C/D frags)
    int n  = wave * 16 + nl;     // global output column (0..127)

    const float* ar0 = As + nl * 260;          // M-subtile 0 (rows 0..15)
    const float* ar1 = As + (16 + nl) * 260;   // M-subtile 1 (rows 16..31)

    v8f acc0 = {}, acc1 = {};
    for (int k = 0; k < Kpad; k += 4) {
        int kb = k + 2 * kh;
        v2f b;
        b.x = Wp[(size_t)kb       * HID + n];
        b.y = Wp[(size_t)(kb + 1) * HID + n];
        v2f a0; a0.x = ar0[kb]; a0.y = ar0[kb + 1];   // -> ds_load_b64
        v2f a1; a1.x = ar1[kb]; a1.y = ar1[kb + 1];
        acc0 = __builtin_amdgcn_wmma_f32_16x16x4_f32(false, a0, false, b,
                                                     (short)0, acc0, false, false);
        acc1 = __builtin_amdgcn_wmma_f32_16x16x4_f32(false, a1, false, b,
                                                     (short)0, acc1, false, false);
    }

    float bn = bias[n];
#pragma unroll
    for (int r = 0; r < 8; r++) {
        int m = r + 8 * kh;                    // C/D layout: VGPR r -> row r (+8 hi lanes)
        float v0v = acc0[r] + bn;
        float v1v = acc1[r] + bn;
        if (relu) { v0v = fmaxf(v0v, 0.f); v1v = fmaxf(v1v, 0.f); }
        out[(size_t)(v0 + m)      * ldo + ocol + n] = v0v;
        out[(size_t)(v0 + 16 + m) * ldo + ocol + n] = v1v;
    }
}

// ---------------------------------------------------------------------------
// acc[e0] += h1[e1]; acc[e1] += h1[e0]  (L2-resident fp32 atomics)
__global__ void scatter_kernel(const float* __restrict__ h1, const int* __restrict__ edges,
                               float* __restrict__ acc) {
    int idx = blockIdx.x * blockDim.x + threadIdx.x;   // e*32 + cq
    int e = idx >> 5;
    if (e >= EN) return;
    int cq = (idx & 31) * 4;
    int ea = edges[2 * e], eb = edges[2 * e + 1];
    float4 va = *(const float4*)(h1 + (size_t)ea * HID + cq);
    float4 vb = *(const float4*)(h1 + (size_t)eb * HID + cq);
    float* da = acc + (size_t)ea * HID + cq;
    float* db = acc + (size_t)eb * HID + cq;
    __hip_atomic_fetch_add(db + 0, va.x, __ATOMIC_RELAXED, __HIP_MEMORY_SCOPE_AGENT);
    __hip_atomic_fetch_add(db + 1, va.y, __ATOMIC_RELAXED, __HIP_MEMORY_SCOPE_AGENT);
    __hip_atomic_fetch_add(db + 2, va.z, __ATOMIC_RELAXED, __HIP_MEMORY_SCOPE_AGENT);
    __hip_atomic_fetch_add(db + 3, va.w, __ATOMIC_RELAXED, __HIP_MEMORY_SCOPE_AGENT);
    __hip_atomic_fetch_add(da + 0, vb.x, __ATOMIC_RELAXED, __HIP_MEMORY_SCOPE_AGENT);
    __hip_atomic_fetch_add(da + 1, vb.y, __ATOMIC_RELAXED, __HIP_MEMORY_SCOPE_AGENT);
    __hip_atomic_fetch_add(da + 2, vb.z, __ATOMIC_RELAXED, __HIP_MEMORY_SCOPE_AGENT);
    __hip_atomic_fetch_add(da + 3, vb.w, __ATOMIC_RELAXED, __HIP_MEMORY_SCOPE_AGENT);
}

// ---------------------------------------------------------------------------
// nopos = relu(acc) -> dst[:,0:128]; dst[:,128:131] = vcur; optional copy into VF259 seg0
__global__ void relu_pack_kernel(const float* __restrict__ acc, const float* __restrict__ vcur,
                                 float* __restrict__ dst, float* __restrict__ dst2) {
    int idx = blockIdx.x * blockDim.x + threadIdx.x;   // v*131 + c
    if (idx >= VN * 131) return;
    int v = idx / 131;
    int c = idx - v * 131;
    if (c < HID) {
        float val = fmaxf(acc[(size_t)v * HID + c], 0.f);
        dst[(size_t)v * 131 + c] = val;
        if (dst2) dst2[(size_t)v * 259 + c] = val;
    } else {
        dst[(size_t)v * 131 + c] = vcur[v * 3 + (c - HID)];
    }
}

__global__ void pos_pack131_kernel(const float* __restrict__ vcur, float* __restrict__ dst) {
    int idx = blockIdx.x * blockDim.x + threadIdx.x;   // v*3 + j
    if (idx >= VN * 3) return;
    int v = idx / 3, j = idx - v * 3;
    dst[(size_t)v * 131 + HID + j] = vcur[idx];
}

// ---------------------------------------------------------------------------
// deform = tanh(vf @ ow + ob); vcur += deform; write outs[s]; forward pos to VF259
__global__ void deform_kernel(const float* __restrict__ vf, const float* __restrict__ ow,
                              const float* __restrict__ ob, float* __restrict__ vcur,
                              float* __restrict__ outp, float* __restrict__ vf259pos) {
    int v = blockIdx.x * blockDim.x + threadIdx.x;
    if (v >= VN) return;
    float d0 = ob[0], d1 = ob[1], d2 = ob[2];
    const float* row = vf + (size_t)v * 131;
    for (int k = 0; k < 131; k++) {
        float a = row[k];
        d0 = fmaf(a, ow[k * 3 + 0], d0);
        d1 = fmaf(a, ow[k * 3 + 1], d1);
        d2 = fmaf(a, ow[k * 3 + 2], d2);
    }
    float p0 = vcur[v * 3 + 0] + tanhf(d0);
    float p1 = vcur[v * 3 + 1] + tanhf(d1);
    float p2 = vcur[v * 3 + 2] + tanhf(d2);
    vcur[v * 3 + 0] = p0; vcur[v * 3 + 1] = p1; vcur[v * 3 + 2] = p2;
    outp[v * 3 + 0] = p0; outp[v * 3 + 1] = p1; outp[v * 3 + 2] = p2;
    if (vf259pos) {
        vf259pos[(size_t)v * 259 + 0] = p0;
        vf259pos[(size_t)v * 259 + 1] = p1;
        vf259pos[(size_t)v * 259 + 2] = p2;
    }
}

// ---------------------------------------------------------------------------
extern "C" void kernel_launch(void* const* d_in, const int* in_sizes, int n_in,
                              void* d_out, int out_size, void* d_ws, size_t ws_size,
                              hipStream_t stream) {
    const float* x     = (const float*)d_in[0];
    const float* verts = (const float*)d_in[1];
    const int*   edges = (const int*)d_in[2];
    const int*   v2m   = (const int*)d_in[3];

    // Param flatten order: jax sorted-keys (bb,bw,[b0,b1,w0,w1]x3,ob,ow) vs
    // insertion order (bw,bb,[w0,b0,w1,b1]x3,ow,ob). Detect via first leaf size.
    bool sorted_order = (n_in > 4) && (in_sizes[4] == HID);
    const float *bbp[3], *bwp[3], *obp[3], *owp[3];
    const float *b0p[3][3], *b1p[3][3], *w0p[3][3], *w1p[3][3];
    for (int s = 0; s < 3; s++) {
        int base = 4 + 16 * s;
        if (sorted_order) {
            bbp[s] = (const float*)d_in[base + 0];
            bwp[s] = (const float*)d_in[base + 1];
            for (int g = 0; g < 3; g++) {
                b0p[s][g] = (const float*)d_in[base + 2 + 4 * g + 0];
                b1p[s][g] = (const float*)d_in[base + 2 + 4 * g + 1];
                w0p[s][g] = (const float*)d_in[base + 2 + 4 * g + 2];
                w1p[s][g] = (const float*)d_in[base + 2 + 4 * g + 3];
            }
            obp[s] = (const float*)d_in[base + 14];
            owp[s] = (const float*)d_in[base + 15];
        } else {
            bwp[s] = (const float*)d_in[base + 0];
            bbp[s] = (const float*)d_in[base + 1];
            for (int g = 0; g < 3; g++) {
                w0p[s][g] = (const float*)d_in[base + 2 + 4 * g + 0];
                b0p[s][g] = (const float*)d_in[base + 2 + 4 * g + 1];
                w1p[s][g] = (const float*)d_in[base + 2 + 4 * g + 2];
                b1p[s][g] = (const float*)d_in[base + 2 + 4 * g + 3];
            }
            owp[s] = (const float*)d_in[base + 14];
            obp[s] = (const float*)d_in[base + 15];
        }
    }

    // Workspace carve-up (floats)
    float* ws = (float*)d_ws;
    size_t off = 0;
    float* xT    = ws + off; off += (size_t)NB * HWD * HWD * CC;   // 0.8M
    float* S     = ws + off; off += (size_t)VN * CC;               // 41.0M
    float* VF259 = ws + off; off += (size_t)VN * 259;              // 41.4M
    float* VF131 = ws + off; off += (size_t)VN * 131;              // 21.0M
    float* accb  = ws + off; off += (size_t)VN * HID;              // 20.5M
    float* h1b   = ws + off; off += (size_t)VN * HID;              // 20.5M
    float* vcur  = ws + off; off += (size_t)VN * 3;                // 0.48M
    // Padded-weight arena (~0.5M floats)
    float* bwpad[3]; float* w0pad[3][3]; float* w1pad[3][3];
    int    w0K[3][3];
    for (int s = 0; s < 3; s++) {
        bwpad[s] = ws + off; off += (size_t)CC * HID;              // K=256 already /4
        for (int g = 0; g < 3; g++) {
            int K = (s > 0 && g == 0) ? 259 : 131;
            int Kpad = (K + 3) & ~3;
            w0K[s][g] = K;
            w0pad[s][g] = ws + off; off += (size_t)Kpad * HID;
            w1pad[s][g] = ws + off; off += (size_t)Kpad * HID;
        }
    }
    (void)ws_size; (void)out_size;

    float* out = (float*)d_out;   // (3, V, 3)

    transpose_x_kernel<<<(NB * CC * HWD * HWD + 255) / 256, 256, 0, stream>>>(x, xT);
    init_pos_kernel<<<(VN * 3 + 255) / 256, 256, 0, stream>>>(verts, vcur);

    // Pre-pad all weight matrices once (tiny)
    for (int s = 0; s < 3; s++) {
        pad_w_kernel<<<(CC * HID + 255) / 256, 256, 0, stream>>>(bwp[s], bwpad[s], CC, CC);
        for (int g = 0; g < 3; g++) {
            int K = w0K[s][g], Kpad = (K + 3) & ~3;
            pad_w_kernel<<<(Kpad * HID + 255) / 256, 256, 0, stream>>>(w0p[s][g], w0pad[s][g], K, Kpad);
            pad_w_kernel<<<(Kpad * HID + 255) / 256, 256, 0, stream>>>(w1p[s][g], w1pad[s][g], K, Kpad);
        }
    }

    const int gemm_blocks    = VN / 32;              // 5000
    const int sample_blocks  = (VN * CC) / 256;      // 160000
    const int scatter_blocks = (EN * 32) / 256;      // 60000
    const int pack_blocks    = (VN * 131 + 255) / 256;

    for (int s = 0; s < 3; s++) {
        // vert_align sample + base linear (fused write into concat buffer)
        sample_kernel<<<sample_blocks, 256, 0, stream>>>(xT, vcur, v2m, S);
        if (s == 0) {
            gemm_wmma_kernel<<<gemm_blocks, 256, 0, stream>>>(
                S, CC, CC, bwpad[s], bbp[s], VF131, 131, 0, 1);
            pos_pack131_kernel<<<(VN * 3 + 255) / 256, 256, 0, stream>>>(vcur, VF131);
        } else {
            // VF259 seg0 (vfeat) and seg2 (pos) were written by previous stage
            gemm_wmma_kernel<<<gemm_blocks, 256, 0, stream>>>(
                S, CC, CC, bwpad[s], bbp[s], VF259, 259, 128, 1);
        }

        for (int g = 0; g < 3; g++) {
            const float* Ain = (s > 0 && g == 0) ? VF259 : VF131;
            int K = w0K[s][g];
            gemm_wmma_kernel<<<gemm_blocks, 256, 0, stream>>>(
                Ain, K, K, w0pad[s][g], b0p[s][g], accb, HID, 0, 0);   // h0 -> acc
            gemm_wmma_kernel<<<gemm_blocks, 256, 0, stream>>>(
                Ain, K, K, w1pad[s][g], b1p[s][g], h1b, HID, 0, 0);    // h1
            scatter_kernel<<<scatter_blocks, 256, 0, stream>>>(h1b, edges, accb);
            float* dst2 = (g == 2 && s < 2) ? VF259 : nullptr;         // vfeat for next stage
            relu_pack_kernel<<<pack_blocks, 256, 0, stream>>>(accb, vcur, VF131, dst2);
        }

        float* vf259pos = (s < 2) ? (VF259 + 256) : nullptr;
        deform_kernel<<<(VN + 255) / 256, 256, 0, stream>>>(
            VF131, owp[s], obp[s], vcur, out + (size_t)s * VN * 3, vf259pos);
    }
}